// MessagePassingLayer_44616120271607
// MI455X (gfx1250) — compile-verified
//
#include <hip/hip_runtime.h>

typedef float v2f __attribute__((ext_vector_type(2)));
typedef float v8f __attribute__((ext_vector_type(8)));

#define DIMC 128          // feature dim D
#define KSTEPS 32         // 128/4 k-steps per input half (H part, E part)
#define NT 8              // 128/16 column tiles
#define LEAKY 0.01f
#define AGG_EPS 1e-7f
#define LN_EPS 1e-5f

// -------------------------------------------------------------------------
// Zero-fill workspace (agg + cnt)
// -------------------------------------------------------------------------
__global__ void zero_kernel(float* __restrict__ p, size_t n) {
    size_t i = (size_t)blockIdx.x * blockDim.x + threadIdx.x;
    if (i < n) p[i] = 0.0f;
}

// -------------------------------------------------------------------------
// Fused edge GEMM + scatter.
// One wave = 16 edges x 128 output cols, K=256 via V_WMMA_F32_16X16X4_F32.
// W (256x128) lives in LDS, pair-swizzled so B fragments are ds_load_b64.
// gather_tail=0: A = [H[head] | E], dst = tail   (msg_fwd)
// gather_tail=1: A = [H[tail] | E], dst = head   (msg_back)
// -------------------------------------------------------------------------
__global__ __launch_bounds__(512) void edge_gemm_scatter(
    const float* __restrict__ H,
    const float* __restrict__ E,
    const int*   __restrict__ ht,
    const float* __restrict__ W,    // [256][128]
    const float* __restrict__ bias, // [128]
    float* __restrict__ agg,        // [n_nodes][128]
    float* __restrict__ cnt,        // [n_nodes]
    int n_edges, int gather_tail)
{
    // pair-swizzled W: ldsW[((r>>1)*128 + c)*2 + (r&1)] = W[r][c]  (128 KB)
    __shared__ float ldsW[2 * DIMC * DIMC];
    for (int i = threadIdx.x; i < 2 * DIMC * DIMC; i += blockDim.x) {
        int r = i >> 7, c = i & 127;
        ldsW[((((unsigned)r >> 1) << 7) + c) * 2 + (r & 1)] = W[i];
    }
    __syncthreads();

    const int lane = threadIdx.x & 31;
    const int l15  = lane & 15;
    const int kh   = lane >> 4;                 // K-half select per ISA A/B layout
    const int wid  = blockIdx.x * (blockDim.x >> 5) + (threadIdx.x >> 5);
    const int nw   = gridDim.x * (blockDim.x >> 5);
    const int ntiles = n_edges >> 4;

    // bias fragment per column tile (loop-invariant)
    float bv[NT];
#pragma unroll
    for (int n = 0; n < NT; ++n) bv[n] = bias[n * 16 + l15];

    const v8f vzero = {0.f, 0.f, 0.f, 0.f, 0.f, 0.f, 0.f, 0.f};

    for (int tile = wid; tile < ntiles; tile += nw) {
        const int e    = (tile << 4) + l15;     // lanes l and l+16 share a row
        const int head = ht[2 * e + 0];
        const int tail = ht[2 * e + 1];
        const int gsrc = gather_tail ? tail : head;
        const int dst  = gather_tail ? head : tail;

        // A-fragment base: row m = l&15, this lane covers K = {2*kh, 2*kh+1} mod 4
        const float* arowH = H + (size_t)gsrc * DIMC + 2 * kh;
        const float* arowE = E + (size_t)e    * DIMC + 2 * kh;

        v8f acc[NT];
#pragma unroll
        for (int n = 0; n < NT; ++n) acc[n] = vzero;

        // ---- K = 0..127 : H-gather half ----
#pragma unroll 4
        for (int k = 0; k < KSTEPS; ++k) {
            v2f a = *(const v2f*)(arowH + 4 * k);
            const float* bp = &ldsW[(2 * k + kh) * (2 * DIMC)];
#pragma unroll
            for (int n = 0; n < NT; ++n) {
                v2f bb = *(const v2f*)(bp + (n * 16 + l15) * 2);
                acc[n] = __builtin_amdgcn_wmma_f32_16x16x4_f32(
                    false, a, false, bb, (short)0, acc[n], false, false);
            }
        }
        // ---- K = 128..255 : E half ----
#pragma unroll 4
        for (int k = 0; k < KSTEPS; ++k) {
            v2f a = *(const v2f*)(arowE + 4 * k);
            const float* bp = &ldsW[(2 * (k + KSTEPS) + kh) * (2 * DIMC)];
#pragma unroll
            for (int n = 0; n < NT; ++n) {
                v2f bb = *(const v2f*)(bp + (n * 16 + l15) * 2);
                acc[n] = __builtin_amdgcn_wmma_f32_16x16x4_f32(
                    false, a, false, bb, (short)0, acc[n], false, false);
            }
        }

        // Epilogue: C layout -> lane l, vgpr r holds row m=(l>>4)*8+r, col 16n+(l&15)
        int dstrow[8];
#pragma unroll
        for (int r = 0; r < 8; ++r)
            dstrow[r] = __shfl(dst, (kh << 3) + r, 32);

#pragma unroll
        for (int n = 0; n < NT; ++n) {
#pragma unroll
            for (int r = 0; r < 8; ++r) {
                atomicAdd(&agg[(size_t)dstrow[r] * DIMC + n * 16 + l15],
                          acc[n][r] + bv[n]);
            }
        }
        if (lane < 16) atomicAdd(&cnt[dst], 1.0f);   // one message per row
    }
}

// -------------------------------------------------------------------------
// Scalar fallback for edges beyond the last full 16-edge tile (n_edges % 16)
// -------------------------------------------------------------------------
__global__ void edge_tail(const float* __restrict__ H, const float* __restrict__ E,
                          const int* __restrict__ ht, const float* __restrict__ W,
                          const float* __restrict__ bias,
                          float* __restrict__ agg, float* __restrict__ cnt,
                          int ebase, int gather_tail)
{
    int e = ebase + blockIdx.x;
    int c = threadIdx.x;                 // 0..127
    int head = ht[2 * e], tail = ht[2 * e + 1];
    int gsrc = gather_tail ? tail : head;
    int dst  = gather_tail ? head : tail;
    const float* hrow = H + (size_t)gsrc * DIMC;
    const float* erow = E + (size_t)e * DIMC;
    float s = bias[c];
    for (int j = 0; j < DIMC; ++j) s += hrow[j] * W[j * DIMC + c];
    for (int j = 0; j < DIMC; ++j) s += erow[j] * W[(DIMC + j) * DIMC + c];
    atomicAdd(&agg[(size_t)dst * DIMC + c], s);
    if (c == 0) atomicAdd(&cnt[dst], 1.0f);
}

// -------------------------------------------------------------------------
// Per-node finalize: mean-divide, LeakyReLU, residual, LayerNorm.
// One wave per node, 4 cols per lane, __shfl_xor wave reductions.
// -------------------------------------------------------------------------
__global__ __launch_bounds__(256) void node_finalize(
    const float* __restrict__ H, const float* __restrict__ agg,
    const float* __restrict__ cnt, const float* __restrict__ gamma,
    const float* __restrict__ beta, float* __restrict__ out, int n_nodes)
{
    int node = blockIdx.x * 8 + (threadIdx.x >> 5);
    if (node >= n_nodes) return;
    int lane = threadIdx.x & 31;

    float4 av = ((const float4*)(agg + (size_t)node * DIMC))[lane];
    float4 hv = ((const float4*)(H   + (size_t)node * DIMC))[lane];
    float inv = 1.0f / (cnt[node] + AGG_EPS);

    float x[4];
    float s = 0.f, q = 0.f;
    float a[4] = {av.x, av.y, av.z, av.w};
    float h[4] = {hv.x, hv.y, hv.z, hv.w};
#pragma unroll
    for (int i = 0; i < 4; ++i) {
        float m = a[i] * inv;
        m = (m >= 0.f) ? m : (LEAKY * m);
        x[i] = m + h[i];
        s += x[i];
        q += x[i] * x[i];
    }
#pragma unroll
    for (int off = 16; off > 0; off >>= 1) {
        s += __shfl_xor(s, off, 32);
        q += __shfl_xor(q, off, 32);
    }
    float mu  = s * (1.0f / DIMC);
    float var = q * (1.0f / DIMC) - mu * mu;
    float rs  = rsqrtf(var + LN_EPS);

    float4 gv = ((const float4*)gamma)[lane];
    float4 bv = ((const float4*)beta)[lane];
    float g[4] = {gv.x, gv.y, gv.z, gv.w};
    float b[4] = {bv.x, bv.y, bv.z, bv.w};
    float4 ov;
    ov.x = g[0] * (x[0] - mu) * rs + b[0];
    ov.y = g[1] * (x[1] - mu) * rs + b[1];
    ov.z = g[2] * (x[2] - mu) * rs + b[2];
    ov.w = g[3] * (x[3] - mu) * rs + b[3];
    ((float4*)(out + (size_t)node * DIMC))[lane] = ov;
}

// -------------------------------------------------------------------------
extern "C" void kernel_launch(void* const* d_in, const int* in_sizes, int n_in,
                              void* d_out, int out_size, void* d_ws, size_t ws_size,
                              hipStream_t stream) {
    const float* H     = (const float*)d_in[0];
    const float* E     = (const float*)d_in[1];
    const int*   ht    = (const int*)d_in[2];
    const float* Wf    = (const float*)d_in[3];
    const float* bf    = (const float*)d_in[4];
    const float* Wb    = (const float*)d_in[5];
    const float* bb    = (const float*)d_in[6];
    const float* gamma = (const float*)d_in[7];
    const float* beta  = (const float*)d_in[8];
    float* out = (float*)d_out;

    const int n_nodes = in_sizes[0] / DIMC;
    const int n_edges = in_sizes[2] / 2;

    float* agg = (float*)d_ws;                       // [n_nodes][128]
    float* cnt = agg + (size_t)n_nodes * DIMC;       // [n_nodes]

    size_t nz = (size_t)n_nodes * DIMC + (size_t)n_nodes;
    zero_kernel<<<(int)((nz + 255) / 256), 256, 0, stream>>>(agg, nz);

    const int grid = 512, blk = 512;                 // 16 waves/WG, 128 KB LDS
    edge_gemm_scatter<<<grid, blk, 0, stream>>>(H, E, ht, Wf, bf, agg, cnt,
                                                n_edges, /*gather_tail=*/0);
    edge_gemm_scatter<<<grid, blk, 0, stream>>>(H, E, ht, Wb, bb, agg, cnt,
                                                n_edges, /*gather_tail=*/1);

    int rem = n_edges & 15;
    if (rem) {
        int ebase = n_edges & ~15;
        edge_tail<<<rem, DIMC, 0, stream>>>(H, E, ht, Wf, bf, agg, cnt, ebase, 0);
        edge_tail<<<rem, DIMC, 0, stream>>>(H, E, ht, Wb, bb, agg, cnt, ebase, 1);
    }

    node_finalize<<<(n_nodes + 7) / 8, 256, 0, stream>>>(H, agg, cnt, gamma, beta,
                                                         out, n_nodes);
}